// SOM_49941879717938
// MI455X (gfx1250) — compile-verified
//
#include <hip/hip_runtime.h>
#include <hip/hip_bf16.h>
#include <stdint.h>

typedef __attribute__((ext_vector_type(2))) float v2f;
typedef __attribute__((ext_vector_type(8))) float v8f;

#define GRID_N 10000
#define GTILES 625     // 625 * 16 = 10000
#define BATCH  128
#define BTILES 8       // 8 * 16 = 128
#define DIM    32

// ws layout (bytes):
//   [0,     1024)  : u64 keys[128]   packed (sortable_dist2_bits<<32 | g)
//   [1024,  41024) : float c2[10000]
//   [41024, 41536) : float x2[128]
//   [41536, 42048) : int  bmu[128]

// ---------------- prep: c2, x2, key init ----------------
__global__ __launch_bounds__(256) void som_prep(
    const float* __restrict__ x, const float* __restrict__ cent,
    float* __restrict__ c2, float* __restrict__ x2,
    unsigned long long* __restrict__ keys) {
  int tid = blockIdx.x * blockDim.x + threadIdx.x;
  if (tid < GRID_N) {
    const float4* row = (const float4*)(cent + (size_t)tid * DIM);
    float s = 0.f;
#pragma unroll
    for (int i = 0; i < DIM / 4; ++i) {
      float4 v = row[i];
      s += v.x * v.x + v.y * v.y + v.z * v.z + v.w * v.w;
    }
    c2[tid] = s;
  } else if (tid < GRID_N + BATCH) {
    int b = tid - GRID_N;
    const float4* row = (const float4*)(x + (size_t)b * DIM);
    float s = 0.f;
#pragma unroll
    for (int i = 0; i < DIM / 4; ++i) {
      float4 v = row[i];
      s += v.x * v.x + v.y * v.y + v.z * v.z + v.w * v.w;
    }
    x2[b] = s;
  } else if (tid < GRID_N + 2 * BATCH) {
    keys[tid - GRID_N - BATCH] = ~0ULL;
  }
}

// ---------------- WMMA distance GEMM + packed argmin ----------------
// D[m][n] = dot(x[m], cent[n]) via 8x V_WMMA_F32_16X16X4_F32.
// argmin over g of dist == argmin over g of (c2[g] - 2*dot) since x2[m] const
// per row and sqrt monotone.
__global__ __launch_bounds__(32) void som_dist_argmin(
    const float* __restrict__ x, const float* __restrict__ cent,
    const float* __restrict__ c2, unsigned long long* __restrict__ keys) {
  const int lane = threadIdx.x;
  const int h = lane >> 4;   // half-wave selector
  const int l = lane & 15;
  const int g0 = blockIdx.x * 16;
  const int m0 = blockIdx.y * 16;

  const float* xrow = x    + (size_t)(m0 + l) * DIM;  // A: M = l
  const float* crow = cent + (size_t)(g0 + l) * DIM;  // B: N = l

  v8f acc = {};
#pragma unroll
  for (int k0 = 0; k0 < DIM; k0 += 4) {
    int k = k0 + 2 * h;
    v2f a; a.x = xrow[k]; a.y = xrow[k + 1];
    v2f b; b.x = crow[k]; b.y = crow[k + 1];
    acc = __builtin_amdgcn_wmma_f32_16x16x4_f32(false, a, false, b,
                                                (short)0, acc, false, false);
  }

  float c2n = c2[g0 + l];
  unsigned int gidx = (unsigned int)(g0 + l);
#pragma unroll
  for (int r = 0; r < 8; ++r) {
    // lane holds D[m = m0 + r + 8h][n = g0 + l]
    float val = c2n - 2.0f * acc[r];
    unsigned int u = __float_as_uint(val);
    u = (u & 0x80000000u) ? ~u : (u | 0x80000000u);  // monotone f32->u32
    unsigned long long key =
        ((unsigned long long)u << 32) | (unsigned long long)gidx;
    // min across the 16 lanes of this half (same m, different g)
#pragma unroll
    for (int s = 1; s < 16; s <<= 1) {
      unsigned long long o = __shfl_xor(key, s, 32);
      if (o < key) key = o;
    }
    if (l == 0) atomicMin(&keys[m0 + r + 8 * h], key);
  }
}

// ---------------- decode keys: bmu_loc, mean mindist ----------------
__global__ __launch_bounds__(128) void som_bmu(
    const unsigned long long* __restrict__ keys, const float* __restrict__ x2,
    const int* __restrict__ locations, float* __restrict__ out,
    int* __restrict__ bmu) {
  __shared__ float red[128];
  int b = threadIdx.x;
  unsigned long long key = keys[b];
  unsigned int gi = (unsigned int)(key & 0xFFFFFFFFu);
  unsigned int u  = (unsigned int)(key >> 32);
  u = (u & 0x80000000u) ? (u & 0x7FFFFFFFu) : ~u;   // inverse monotone map
  float val = __uint_as_float(u);
  float md = sqrtf(fmaxf(x2[b] + val, 0.0f));
  bmu[b] = (int)gi;
  out[2 * b]     = (float)locations[2 * gi];
  out[2 * b + 1] = (float)locations[2 * gi + 1];
  red[b] = md;
  __syncthreads();
  for (int s = 64; s > 0; s >>= 1) {
    if (b < s) red[b] += red[b + s];
    __syncthreads();
  }
  if (b == 0) out[256] = red[0] * (1.0f / (float)BATCH);
}

// ---------------- WMMA update GEMM: T = exp(-dm/den)^T @ X ----------------
// new_w[g,d] = w + (alpha*lr/B) * (T[g,d] - s[g]*w),  s[g] = sum_b exp(.)
__global__ __launch_bounds__(32) void som_update(
    const float* __restrict__ x, const float* __restrict__ cent,
    const float* __restrict__ dm, const int* __restrict__ bmu,
    const float* __restrict__ lr_p, float* __restrict__ out_w) {
  const int lane = threadIdx.x;
  const int h = lane >> 4;
  const int l = lane & 15;
  const int g0 = blockIdx.x * 16;

  float lr = lr_p[0];
  float sigma_op = 50.0f * lr;                          // SIGMA = 50
  float invDen = 1.0f / (2.0f * sigma_op * sigma_op + 1e-5f);
  float scale = (0.05f * lr) * (1.0f / (float)BATCH);   // alpha_op / B

  v8f acc0 = {}, acc1 = {};
  float ssum = 0.0f;
#pragma unroll 4
  for (int k0 = 0; k0 < BATCH; k0 += 4) {
    int b0 = k0 + 2 * h;
    int i0 = bmu[b0], i1 = bmu[b0 + 1];
    // A[M = g_local = l][K = b0/b0+1] = exp(-dm[bmu[b]*G + g] * invDen)
    float a0 = expf(-dm[(size_t)i0 * GRID_N + g0 + l] * invDen);
    float a1 = expf(-dm[(size_t)i1 * GRID_N + g0 + l] * invDen);
    v2f a; a.x = a0; a.y = a1;
    ssum += a0 + a1;
    // B[K = b][N = d] = x[b][d]; two N-tiles cover DIM=32
    v2f bb0; bb0.x = x[(size_t)b0 * DIM + l];
             bb0.y = x[(size_t)(b0 + 1) * DIM + l];
    v2f bb1; bb1.x = x[(size_t)b0 * DIM + 16 + l];
             bb1.y = x[(size_t)(b0 + 1) * DIM + 16 + l];
    acc0 = __builtin_amdgcn_wmma_f32_16x16x4_f32(false, a, false, bb0,
                                                 (short)0, acc0, false, false);
    acc1 = __builtin_amdgcn_wmma_f32_16x16x4_f32(false, a, false, bb1,
                                                 (short)0, acc1, false, false);
  }
  // s for g = g0 + (lane&15), replicated to all lanes
  float stot = ssum + __shfl_xor(ssum, 16, 32);
#pragma unroll
  for (int r = 0; r < 8; ++r) {
    int gl = r + 8 * h;                       // row of D this lane holds
    float s_r = __shfl(stot, gl, 32);         // s[g0 + gl]
    size_t row = (size_t)(g0 + gl) * DIM;
    float w0 = cent[row + l];
    float w1 = cent[row + 16 + l];
    out_w[row + l]      = w0 + scale * (acc0[r] - s_r * w0);
    out_w[row + 16 + l] = w1 + scale * (acc1[r] - s_r * w1);
  }
}

extern "C" void kernel_launch(void* const* d_in, const int* in_sizes, int n_in,
                              void* d_out, int out_size, void* d_ws,
                              size_t ws_size, hipStream_t stream) {
  const float* x    = (const float*)d_in[0];   // [128,32]
  const float* cent = (const float*)d_in[1];   // [10000,32]
  const int*   loc  = (const int*)d_in[2];     // [10000,2]
  const float* dm   = (const float*)d_in[3];   // [10000,10000]
  const float* lr   = (const float*)d_in[4];   // [1]
  float* out = (float*)d_out;                  // 256 + 1 + 320000

  char* ws = (char*)d_ws;
  unsigned long long* keys = (unsigned long long*)ws;
  float* c2 = (float*)(ws + 1024);
  float* x2 = (float*)(ws + 1024 + 40000);
  int*   bmu = (int*)(ws + 1024 + 40000 + 512);

  som_prep<<<(GRID_N + 2 * BATCH + 255) / 256, 256, 0, stream>>>(
      x, cent, c2, x2, keys);
  som_dist_argmin<<<dim3(GTILES, BTILES), 32, 0, stream>>>(x, cent, c2, keys);
  som_bmu<<<1, 128, 0, stream>>>(keys, x2, loc, out, bmu);
  som_update<<<GTILES, 32, 0, stream>>>(x, cent, dm, bmu, lr, out + 257);
}